// Attn_29815662969013
// MI455X (gfx1250) — compile-verified
//
#include <hip/hip_runtime.h>
#include <math.h>

typedef __attribute__((ext_vector_type(2))) float v2f;
typedef __attribute__((ext_vector_type(4))) float f4;
typedef __attribute__((ext_vector_type(8))) float v8f;

#define H 1024
#define S 32768

// workspace layout (float offsets)
#define WS_PARTIAL 0      // 16*1024 floats
#define WS_U       16384  // 1024 floats
#define WS_BMAX    17408  // 256 floats
#define WS_GMAX    17664  // 1 float
#define WS_BSUM    17665  // 32 floats
#define WS_INV     17697  // 1 float

// ---- K1: partial u over d-chunks: partial[c*H + h] = sum_{d in chunk c} W[d,h]*hidden[d]
__global__ void __launch_bounds__(256)
k_partial_u(const float* __restrict__ W, const float* __restrict__ hidden,
            float* __restrict__ partial) {
    const int h4 = threadIdx.x * 4;            // 256 threads * 4 cols = 1024 cols
    const int d0 = blockIdx.x * 64;            // 16 blocks * 64 rows = 1024 rows
    f4 acc = {0.f, 0.f, 0.f, 0.f};
    #pragma unroll 8
    for (int d = d0; d < d0 + 64; ++d) {
        f4 w = *(const f4*)(W + (size_t)d * H + h4);
        acc += w * hidden[d];
    }
    *(f4*)(partial + blockIdx.x * H + h4) = acc;
}

// ---- K2: u[h] = sum over 16 partials
__global__ void __launch_bounds__(256)
k_reduce_u(const float* __restrict__ partial, float* __restrict__ u) {
    const int h4 = threadIdx.x * 4;
    f4 acc = {0.f, 0.f, 0.f, 0.f};
    #pragma unroll
    for (int j = 0; j < 16; ++j)
        acc += *(const f4*)(partial + j * H + h4);
    *(f4*)(u + h4) = acc;
}

// ---- K3: energies e[s] = enc[s,:].u via V_WMMA_F32_16X16X4_F32, plus per-block max
__global__ void __launch_bounds__(256)
k_energies(const float* __restrict__ enc, const float* __restrict__ u,
           float* __restrict__ e_out, float* __restrict__ bmax) {
    const int tid  = threadIdx.x;
    const int wave = tid >> 5;
    const int lane = tid & 31;
    const int half = lane >> 4;   // 0: K=0,1 slots ; 1: K=2,3 slots
    const int row  = lane & 15;
    const int s_base = (blockIdx.x * 8 + wave) * 16;

    const float* rowp = enc + (size_t)(s_base + row) * H + 4 * half;
    const float* up   = u + 4 * half;

    v8f acc = {0.f, 0.f, 0.f, 0.f, 0.f, 0.f, 0.f, 0.f};

    #pragma unroll 4
    for (int k8 = 0; k8 < H; k8 += 8) {
        // per lane: 4 consecutive enc floats + matching 4 u floats.
        // WMMA#0 consumes elements {0,1}, WMMA#1 consumes {2,3}; each (lane,vgpr)
        // slot pairs enc[row][k] with u[k] for the same k, so the K-sum is exact.
        f4 av = __builtin_nontemporal_load((const f4*)(rowp + k8));
        f4 bv = *(const f4*)(up + k8);
        v2f a0 = {av.x, av.y}, a1 = {av.z, av.w};
        v2f b0 = {bv.x, bv.y}, b1 = {bv.z, bv.w};
        acc = __builtin_amdgcn_wmma_f32_16x16x4_f32(false, a0, false, b0,
                                                    (short)0, acc, false, false);
        acc = __builtin_amdgcn_wmma_f32_16x16x4_f32(false, a1, false, b1,
                                                    (short)0, acc, false, false);
    }

    // D layout: lanes 0-15 vgpr j = e[s_base+j]; lanes 16-31 vgpr j = e[s_base+8+j]
    // (all 16 columns identical since B was replicated). Lane 0 / lane 16 write.
    if (row == 0) {
        float* dst = e_out + s_base + 8 * half;
        f4 lo = {acc[0], acc[1], acc[2], acc[3]};
        f4 hi = {acc[4], acc[5], acc[6], acc[7]};
        *(f4*)(dst)     = lo;
        *(f4*)(dst + 4) = hi;
    }

    // wave max -> block max
    float m = acc[0];
    #pragma unroll
    for (int j = 1; j < 8; ++j) m = fmaxf(m, acc[j]);
    m = fmaxf(m, __shfl_xor(m, 16, 32));   // combine halves (rows 0-7 vs 8-15)

    __shared__ float smax[8];
    if (lane == 0) smax[wave] = m;
    __syncthreads();
    if (tid == 0) {
        float bm = smax[0];
        #pragma unroll
        for (int j = 1; j < 8; ++j) bm = fmaxf(bm, smax[j]);
        bmax[blockIdx.x] = bm;
    }
}

// ---- K4: global max over 256 block maxima
__global__ void __launch_bounds__(256)
k_gmax(const float* __restrict__ bmax, float* __restrict__ gmax) {
    __shared__ float sm[256];
    sm[threadIdx.x] = bmax[threadIdx.x];
    __syncthreads();
    for (int off = 128; off > 0; off >>= 1) {
        if (threadIdx.x < off)
            sm[threadIdx.x] = fmaxf(sm[threadIdx.x], sm[threadIdx.x + off]);
        __syncthreads();
    }
    if (threadIdx.x == 0) *gmax = sm[0];
}

// ---- K5: e = exp(e - gmax), per-block partial sums (32 blocks x 256 thr x 4 elems)
__global__ void __launch_bounds__(256)
k_exp(float* __restrict__ e, const float* __restrict__ gmax,
      float* __restrict__ bsum) {
    const float M = *gmax;
    const int i = (blockIdx.x * 256 + threadIdx.x) * 4;
    f4 v = *(f4*)(e + i);
    v.x = expf(v.x - M);
    v.y = expf(v.y - M);
    v.z = expf(v.z - M);
    v.w = expf(v.w - M);
    *(f4*)(e + i) = v;

    __shared__ float ss[256];
    ss[threadIdx.x] = v.x + v.y + v.z + v.w;
    __syncthreads();
    for (int off = 128; off > 0; off >>= 1) {
        if (threadIdx.x < off) ss[threadIdx.x] += ss[threadIdx.x + off];
        __syncthreads();
    }
    if (threadIdx.x == 0) bsum[blockIdx.x] = ss[0];
}

// ---- K6: total sum -> 1/sum (single wave)
__global__ void __launch_bounds__(32)
k_sum(const float* __restrict__ bsum, float* __restrict__ inv) {
    float s = bsum[threadIdx.x];
    for (int off = 16; off > 0; off >>= 1) s += __shfl_down(s, off, 32);
    if (threadIdx.x == 0) *inv = 1.0f / s;
}

// ---- K7: scale
__global__ void __launch_bounds__(256)
k_scale(float* __restrict__ e, const float* __restrict__ inv) {
    const float r = *inv;
    const int i = (blockIdx.x * 256 + threadIdx.x) * 4;
    f4 v = *(f4*)(e + i);
    v *= r;
    *(f4*)(e + i) = v;
}

extern "C" void kernel_launch(void* const* d_in, const int* in_sizes, int n_in,
                              void* d_out, int out_size, void* d_ws, size_t ws_size,
                              hipStream_t stream) {
    const float* hidden = (const float*)d_in[0];   // [H]
    const float* enc    = (const float*)d_in[1];   // [S,H]
    const float* W      = (const float*)d_in[2];   // [H,H]
    // d_in[3] = b : mathematically irrelevant (softmax shift invariance)
    float* ws  = (float*)d_ws;
    float* out = (float*)d_out;                    // [S]

    k_partial_u<<<16, 256, 0, stream>>>(W, hidden, ws + WS_PARTIAL);
    k_reduce_u <<<1, 256, 0, stream>>>(ws + WS_PARTIAL, ws + WS_U);
    k_energies <<<S / 128, 256, 0, stream>>>(enc, ws + WS_U, out, ws + WS_BMAX);
    k_gmax     <<<1, 256, 0, stream>>>(ws + WS_BMAX, ws + WS_GMAX);
    k_exp      <<<32, 256, 0, stream>>>(out, ws + WS_GMAX, ws + WS_BSUM);
    k_sum      <<<1, 32, 0, stream>>>(ws + WS_BSUM, ws + WS_INV);
    k_scale    <<<32, 256, 0, stream>>>(out, ws + WS_INV);
}